// FarthestPointSampling_26980984553513
// MI455X (gfx1250) — compile-verified
//
#include <hip/hip_runtime.h>
#include <stdint.h>

#define NB    32          // batches (blocks)
#define NPT   131072      // points per batch
#define KSEL  1024        // samples
#define INF_F 1e10f

typedef float v4f __attribute__((ext_vector_type(4)));

// ---------------------------------------------------------------------------
// Kernel 1 (fast path): SoA + packed f32; dist lives in LDS (38 quads/thread)
// + global ws (26 quads/thread). No register-resident dist => no spills.
// ---------------------------------------------------------------------------
#define NTQ   512
#define CHL   38          // dist quads per thread in LDS
#define CHG   26          // dist quads per thread in global ws (38+26 = 64 = NPT/NTQ/4)
#define SOA_BYTES   ((size_t)NB * NPT * 3 * sizeof(float))        // 48 MB
#define DIST_BYTES  ((size_t)NB * CHG * NTQ * 16)                 // 6.8 MB

__global__ __launch_bounds__(NTQ, 1)
void fps_soa_kernel(const float* __restrict__ pts,   // original AoS (centroid/gather)
                    const float* __restrict__ W,     // SoA workspace
                    v4f*         __restrict__ D,     // global dist quads
                    const int*   __restrict__ init_idx,
                    float*       __restrict__ out)
{
    #pragma clang fp contract(off)
    __shared__ v4f                s_dist4[CHL * NTQ];    // 311296 B
    __shared__ int                s_chosen[KSEL];
    __shared__ float              s_cent[4];
    __shared__ unsigned long long s_key[2];

    const int tid = threadIdx.x;
    const int b   = blockIdx.x;
    const float* __restrict__ pb = pts + (size_t)b * ((size_t)NPT * 3);
    const float* __restrict__ Xb = W + (size_t)b * (3 * (size_t)NPT);
    const float* __restrict__ YA = Xb + NPT + 4 * (size_t)tid;
    const float* __restrict__ ZA = Xb + 2 * NPT + 4 * (size_t)tid;
    const float* __restrict__ XA = Xb + 4 * (size_t)tid;
    // quad code c (0..63): quad q = c*512 + tid, points 4q..4q+3,
    // element base offset = c*2048 + 4*tid floats
    v4f* __restrict__ DA = D + ((size_t)b * CHG * NTQ) + tid;   // + cc*NTQ per quad

    if (tid == 0) {
        s_key[0] = 0ULL;
        s_key[1] = 0ULL;
        int idx0 = init_idx[b];
        s_chosen[0] = idx0;
        const float* cp = pb + 3 * (size_t)idx0;
        uint32_t lc = (uint32_t)(uintptr_t)(&s_cent[0]);
        asm volatile(
            "global_load_async_to_lds_b32 %0, %1, off\n\t"
            "global_load_async_to_lds_b32 %0, %1, off offset:4\n\t"
            "global_load_async_to_lds_b32 %0, %1, off offset:8\n\t"
            "s_wait_asynccnt 0"
            :: "v"(lc), "v"(cp) : "memory");
    }

    const v4f inf4 = {INF_F, INF_F, INF_F, INF_F};
    #pragma unroll
    for (int c = 0; c < CHL; ++c) s_dist4[c * NTQ + tid] = inf4;
    #pragma unroll
    for (int cc = 0; cc < CHG; ++cc) DA[cc * NTQ] = inf4;

    __syncthreads();

    #pragma unroll 1
    for (int k = 0; k < KSEL - 1; ++k) {
        const float cx = s_cent[0];
        const float cy = s_cent[1];
        const float cz = s_cent[2];
        const v4f cx4 = {cx, cx, cx, cx};
        const v4f cy4 = {cy, cy, cy, cy};
        const v4f cz4 = {cz, cz, cz, cz};

        float bv0 = -1.0f, bv1 = -1.0f, bv2 = -1.0f, bv3 = -1.0f;
        int   bc0 = 0,     bc1 = 0,     bc2 = 0,     bc3 = 0;

        // part 1: dist quads in LDS, codes 0..CHL-1
        #pragma unroll
        for (int cc0 = 0; cc0 < CHL; cc0 += 4) {
            #pragma unroll
            for (int c = cc0; c < (cc0 + 4 < CHL ? cc0 + 4 : CHL); ++c) {
                v4f xx = *(const v4f*)(XA + 2048 * (size_t)c);
                v4f yy = *(const v4f*)(YA + 2048 * (size_t)c);
                v4f zz = *(const v4f*)(ZA + 2048 * (size_t)c);
                v4f dx = xx - cx4, dy = yy - cy4, dz = zz - cz4;
                v4f d2 = dx * dx + dy * dy + dz * dz;   // contract(off): matches ref
                v4f old = s_dist4[c * NTQ + tid];
                v4f nd;
                nd.x = fminf(old.x, d2.x);
                nd.y = fminf(old.y, d2.y);
                nd.z = fminf(old.z, d2.z);
                nd.w = fminf(old.w, d2.w);
                s_dist4[c * NTQ + tid] = nd;
                if (nd.x > bv0) { bv0 = nd.x; bc0 = c; }
                if (nd.y > bv1) { bv1 = nd.y; bc1 = c; }
                if (nd.z > bv2) { bv2 = nd.z; bc2 = c; }
                if (nd.w > bv3) { bv3 = nd.w; bc3 = c; }
            }
            __builtin_amdgcn_sched_barrier(0);
        }

        // part 2: dist quads in global ws, codes CHL..CHL+CHG-1
        #pragma unroll
        for (int cc0 = 0; cc0 < CHG; cc0 += 4) {
            #pragma unroll
            for (int cc = cc0; cc < (cc0 + 4 < CHG ? cc0 + 4 : CHG); ++cc) {
                const int c = CHL + cc;
                v4f xx = *(const v4f*)(XA + 2048 * (size_t)c);
                v4f yy = *(const v4f*)(YA + 2048 * (size_t)c);
                v4f zz = *(const v4f*)(ZA + 2048 * (size_t)c);
                v4f dx = xx - cx4, dy = yy - cy4, dz = zz - cz4;
                v4f d2 = dx * dx + dy * dy + dz * dz;
                v4f old = DA[cc * NTQ];
                v4f nd;
                nd.x = fminf(old.x, d2.x);
                nd.y = fminf(old.y, d2.y);
                nd.z = fminf(old.z, d2.z);
                nd.w = fminf(old.w, d2.w);
                DA[cc * NTQ] = nd;
                if (nd.x > bv0) { bv0 = nd.x; bc0 = c; }
                if (nd.y > bv1) { bv1 = nd.y; bc1 = c; }
                if (nd.z > bv2) { bv2 = nd.z; bc2 = c; }
                if (nd.w > bv3) { bv3 = nd.w; bc3 = c; }
            }
            __builtin_amdgcn_sched_barrier(0);
        }

        // point index n = (code*512 + tid)*4 + element; key = (bits(dist)<<32) | ~n
        const int tb = tid << 2;
        unsigned long long k0 = ((unsigned long long)__float_as_uint(bv0) << 32) |
                                (unsigned long long)(uint32_t)(~((bc0 << 11) + tb + 0));
        unsigned long long k1 = ((unsigned long long)__float_as_uint(bv1) << 32) |
                                (unsigned long long)(uint32_t)(~((bc1 << 11) + tb + 1));
        unsigned long long k2 = ((unsigned long long)__float_as_uint(bv2) << 32) |
                                (unsigned long long)(uint32_t)(~((bc2 << 11) + tb + 2));
        unsigned long long k3 = ((unsigned long long)__float_as_uint(bv3) << 32) |
                                (unsigned long long)(uint32_t)(~((bc3 << 11) + tb + 3));
        unsigned long long key = k0 > k1 ? k0 : k1;
        unsigned long long kB  = k2 > k3 ? k2 : k3;
        key = key > kB ? key : kB;

        #pragma unroll
        for (int m = 16; m >= 1; m >>= 1) {
            unsigned long long o = __shfl_xor(key, m, 32);
            key = (o > key) ? o : key;
        }
        if ((tid & 31) == 0) atomicMax(&s_key[k & 1], key);  // ds_max_u64, 1/wave
        __syncthreads();                                      // barrier 1

        unsigned long long w = s_key[k & 1];
        int idx = (int)(~(uint32_t)w);
        if (tid == 0) {
            s_chosen[k + 1]    = idx;
            s_key[(k + 1) & 1] = 0ULL;
            const float* cp = pb + 3 * (size_t)idx;
            uint32_t lc = (uint32_t)(uintptr_t)(&s_cent[0]);
            asm volatile(
                "global_load_async_to_lds_b32 %0, %1, off\n\t"
                "global_load_async_to_lds_b32 %0, %1, off offset:4\n\t"
                "global_load_async_to_lds_b32 %0, %1, off offset:8\n\t"
                "s_wait_asynccnt 0"
                :: "v"(lc), "v"(cp) : "memory");
        }
        __syncthreads();                                      // barrier 2
    }

    #pragma unroll
    for (int t = tid; t < KSEL; t += NTQ) {
        int idx = s_chosen[t];
        float x = pb[3 * (size_t)idx + 0];
        float y = pb[3 * (size_t)idx + 1];
        float z = pb[3 * (size_t)idx + 2];
        float* ob = out + ((size_t)b * KSEL + t) * 3;
        ob[0] = x; ob[1] = y; ob[2] = z;
    }
}

// ---------------------------------------------------------------------------
// Kernel 0: deinterleave AoS xyz -> per-batch SoA X[N], Y[N], Z[N] in d_ws
// ---------------------------------------------------------------------------
__global__ __launch_bounds__(1024, 1)
void deint_kernel(const float* __restrict__ pts, float* __restrict__ W)
{
    int g = blockIdx.x * 1024 + threadIdx.x;        // 0 .. NB*NPT-1
    int b = g >> 17;                                // NPT = 2^17
    int n = g & (NPT - 1);
    float x = pts[3 * (size_t)g + 0];
    float y = pts[3 * (size_t)g + 1];
    float z = pts[3 * (size_t)g + 2];
    float* Wb = W + (size_t)b * (3 * (size_t)NPT);
    Wb[n]           = x;
    Wb[NPT + n]     = y;
    Wb[2 * NPT + n] = z;
}

// ---------------------------------------------------------------------------
// Kernel 2 (fallback if ws too small): proven AoS kernel (round-4 version)
// ---------------------------------------------------------------------------
#define NT    512
#define CHA   104
#define CHB   152

__global__ __launch_bounds__(NT, 1)
void fps_aos_kernel(const float* __restrict__ points,
                    const int* __restrict__ init_idx,
                    float* __restrict__ out)
{
    __shared__ float              s_dist[CHB * NT];
    __shared__ int                s_chosen[KSEL];
    __shared__ float              s_cent[4];
    __shared__ unsigned long long s_key[2];

    const int tid = threadIdx.x;
    const int b   = blockIdx.x;
    const float* __restrict__ pb = points + (size_t)b * ((size_t)NPT * 3);
    const float* __restrict__ pA = pb + 3 * (size_t)tid;
    const float* __restrict__ pB = pb + 3 * ((size_t)CHA * NT + tid);

    if (tid == 0) {
        s_key[0] = 0ULL;
        s_key[1] = 0ULL;
        int idx0 = init_idx[b];
        s_chosen[0] = idx0;
        const float* cp = pb + 3 * (size_t)idx0;
        uint32_t lc = (uint32_t)(uintptr_t)(&s_cent[0]);
        asm volatile(
            "global_load_async_to_lds_b32 %0, %1, off\n\t"
            "global_load_async_to_lds_b32 %0, %1, off offset:4\n\t"
            "global_load_async_to_lds_b32 %0, %1, off offset:8\n\t"
            "s_wait_asynccnt 0"
            :: "v"(lc), "v"(cp) : "memory");
    }

    float distA[CHA];
    #pragma unroll
    for (int j = 0; j < CHA; ++j) distA[j] = INF_F;
    #pragma unroll
    for (int j = 0; j < CHB; ++j) s_dist[j * NT + tid] = INF_F;

    __syncthreads();

    #pragma unroll 1
    for (int k = 0; k < KSEL - 1; ++k) {
        const float cx = s_cent[0];
        const float cy = s_cent[1];
        const float cz = s_cent[2];
        float bestVal  = -1.0f;
        int   bestCode = 0;

        #pragma unroll
        for (int j = 0; j < CHA; ++j) {
            float x = pA[3 * NT * j + 0];
            float y = pA[3 * NT * j + 1];
            float z = pA[3 * NT * j + 2];
            float dx = x - cx, dy = y - cy, dz = z - cz;
            float d2 = __fadd_rn(__fadd_rn(__fmul_rn(dx, dx), __fmul_rn(dy, dy)),
                                 __fmul_rn(dz, dz));
            float nd = fminf(distA[j], d2);
            distA[j] = nd;
            if (nd > bestVal) { bestVal = nd; bestCode = j; }
        }
        #pragma unroll
        for (int j = 0; j < CHB; ++j) {
            float x = pB[3 * NT * j + 0];
            float y = pB[3 * NT * j + 1];
            float z = pB[3 * NT * j + 2];
            float dx = x - cx, dy = y - cy, dz = z - cz;
            float d2 = __fadd_rn(__fadd_rn(__fmul_rn(dx, dx), __fmul_rn(dy, dy)),
                                 __fmul_rn(dz, dz));
            float nd = fminf(s_dist[j * NT + tid], d2);
            s_dist[j * NT + tid] = nd;
            if (nd > bestVal) { bestVal = nd; bestCode = CHA + j; }
        }

        const int bestN = (bestCode << 9) + tid;
        unsigned long long key =
            ((unsigned long long)__float_as_uint(bestVal) << 32) |
            (unsigned long long)(uint32_t)(~bestN);
        #pragma unroll
        for (int m = 16; m >= 1; m >>= 1) {
            unsigned long long o = __shfl_xor(key, m, 32);
            key = (o > key) ? o : key;
        }
        if ((tid & 31) == 0) atomicMax(&s_key[k & 1], key);
        __syncthreads();
        unsigned long long w = s_key[k & 1];
        int idx = (int)(~(uint32_t)w);
        if (tid == 0) {
            s_chosen[k + 1]    = idx;
            s_key[(k + 1) & 1] = 0ULL;
            const float* cp = pb + 3 * (size_t)idx;
            uint32_t lc = (uint32_t)(uintptr_t)(&s_cent[0]);
            asm volatile(
                "global_load_async_to_lds_b32 %0, %1, off\n\t"
                "global_load_async_to_lds_b32 %0, %1, off offset:4\n\t"
                "global_load_async_to_lds_b32 %0, %1, off offset:8\n\t"
                "s_wait_asynccnt 0"
                :: "v"(lc), "v"(cp) : "memory");
        }
        __syncthreads();
    }

    #pragma unroll
    for (int t = tid; t < KSEL; t += NT) {
        int idx = s_chosen[t];
        float x = pb[3 * (size_t)idx + 0];
        float y = pb[3 * (size_t)idx + 1];
        float z = pb[3 * (size_t)idx + 2];
        float* ob = out + ((size_t)b * KSEL + t) * 3;
        ob[0] = x; ob[1] = y; ob[2] = z;
    }
}

extern "C" void kernel_launch(void* const* d_in, const int* in_sizes, int n_in,
                              void* d_out, int out_size, void* d_ws, size_t ws_size,
                              hipStream_t stream) {
    const float* points   = (const float*)d_in[0];
    const int*   init_idx = (const int*)d_in[1];
    float*       out      = (float*)d_out;
    (void)in_sizes; (void)n_in; (void)out_size;

    const size_t need = SOA_BYTES + DIST_BYTES;   // ~57 MB
    if (ws_size >= need) {
        float* W = (float*)d_ws;
        v4f*   D = (v4f*)((char*)d_ws + SOA_BYTES);
        deint_kernel<<<dim3((NB * NPT) / 1024), dim3(1024), 0, stream>>>(points, W);
        fps_soa_kernel<<<dim3(NB), dim3(NTQ), 0, stream>>>(points, W, D, init_idx, out);
    } else {
        fps_aos_kernel<<<dim3(NB), dim3(NT), 0, stream>>>(points, init_idx, out);
    }
}